// Magnus2nd_80762565034565
// MI455X (gfx1250) — compile-verified
//
#include <hip/hip_runtime.h>
#include <hip/hip_bf16.h>

typedef __attribute__((ext_vector_type(2))) float v2f;
typedef __attribute__((ext_vector_type(4))) float v4f;
typedef __attribute__((ext_vector_type(8))) float v8f;

#define DIM   64
#define LSTR  68          // padded LDS row stride: bank = (4*i + j) % 64 -> conflict-free
#define NTHREADS 256      // 8 waves (wave32)

// One 64x64x64 fp32 matmul stage on the WMMA pipe.
//   Arow : left matrix  A, row-major  (A[m][k] at m*LSTR+k)
//   Bcol : right matrix B, col-major  (B[k][n] at n*LSTR+k)
//   result = scale * (A @ B) [+ I]
//   OutC (optional): col-major result (result[m][n] at n*LSTR+m)  -> 2x b128 per tile
//   OutR (optional): row-major result                              -> 16x b32 per wave
// 16 C-tiles of 16x16; wave w owns tiles (mt = w>>1, nt0 = (w&1)*2 .. +1).
// K accumulated via 16 chained V_WMMA_F32_16X16X4_F32.
__device__ __forceinline__ void wmma_stage(const float* Arow,
                                           const float* Bcol,
                                           float* OutC,
                                           float* OutR,
                                           float scale, bool addI)
{
    const int tid  = threadIdx.x;
    const int wave = tid >> 5;
    const int lane = tid & 31;
    const int half = lane >> 4;   // 0: lanes 0-15, 1: lanes 16-31
    const int l15  = lane & 15;
    const int mt   = wave >> 1;         // C-tile row block (0..3)
    const int nt0  = (wave & 1) * 2;    // first of two C-tile col blocks

    v8f c0 = {}; v8f c1 = {};

    // A 16x4 f32 frag (ISA 7.12.2): VGPR0={K0|K2}, VGPR1={K1|K3}; row-major -> contiguous b64
    const float* arow  = Arow + (mt * 16 + l15) * LSTR + half * 2;
    // B 4x16 f32 frag: VGPR0=row(k+2h), VGPR1=row(k+2h+1) at col n; col-major -> contiguous b64
    const float* bcol0 = Bcol + (nt0 * 16 + l15) * LSTR + half * 2;
    const float* bcol1 = bcol0 + 16 * LSTR;

    #pragma unroll
    for (int k = 0; k < DIM; k += 4) {
        const v2f a  = *reinterpret_cast<const v2f*>(arow  + k);
        const v2f b0 = *reinterpret_cast<const v2f*>(bcol0 + k);
        const v2f b1 = *reinterpret_cast<const v2f*>(bcol1 + k);
        c0 = __builtin_amdgcn_wmma_f32_16x16x4_f32(false, a, false, b0,
                                                   (short)0, c0, false, false);
        c1 = __builtin_amdgcn_wmma_f32_16x16x4_f32(false, a, false, b1,
                                                   (short)0, c1, false, false);
    }

    // Scale + optional identity. C/D layout: VGPR j -> row (mt*16 + 8*half + j), col fixed.
    const int col0 = nt0 * 16 + l15;
    const int col1 = col0 + 16;
    #pragma unroll
    for (int j = 0; j < 8; ++j) {
        const int row = mt * 16 + half * 8 + j;
        c0[j] = scale * c0[j] + ((addI && row == col0) ? 1.0f : 0.0f);
        c1[j] = scale * c1[j] + ((addI && row == col1) ? 1.0f : 0.0f);
    }

    if (OutC) {
        // per lane: 8 consecutive rows of one column -> contiguous col-major, 16B aligned
        float* p0 = OutC + col0 * LSTR + mt * 16 + half * 8;
        float* p1 = p0 + 16 * LSTR;
        *reinterpret_cast<v4f*>(p0)     = (v4f){c0[0], c0[1], c0[2], c0[3]};
        *reinterpret_cast<v4f*>(p0 + 4) = (v4f){c0[4], c0[5], c0[6], c0[7]};
        *reinterpret_cast<v4f*>(p1)     = (v4f){c1[0], c1[1], c1[2], c1[3]};
        *reinterpret_cast<v4f*>(p1 + 4) = (v4f){c1[4], c1[5], c1[6], c1[7]};
    }
    if (OutR) {
        #pragma unroll
        for (int j = 0; j < 8; ++j) {
            const int row = mt * 16 + half * 8 + j;
            OutR[row * LSTR + col0] = c0[j];
            OutR[row * LSTR + col1] = c1[j];
        }
    }
}

// Magnus 2nd order step: y = expm(h*A) @ y0, one workgroup per batch element.
// expm via scaling-and-squaring + degree-8 Taylor (Horner), all matmuls on WMMA.
__global__ __launch_bounds__(NTHREADS)
void magnus2_expm_kernel(const float* __restrict__ A,
                         const float* __restrict__ h,
                         const float* __restrict__ y0,
                         float* __restrict__ yout)
{
    __shared__ float Xs[DIM * LSTR];   // row-major X; reused as row-major ping after Horner
    __shared__ float R0[DIM * LSTR];   // row-major T
    __shared__ float C0[DIM * LSTR];   // col-major T ping
    __shared__ float C1[DIM * LSTR];   // col-major T pong
    __shared__ float rowsum[DIM];
    __shared__ float y0s[DIM];
    __shared__ float psum[4 * DIM];

    const int b   = blockIdx.x;
    const int tid = threadIdx.x;
    const float hb = h[b];

    // ---- Load A (64x64 fp32, 16 KB) coalesced as float4, store raw into Xs ----
    const float* Ab = A + (size_t)b * DIM * DIM;
    #pragma unroll
    for (int i = 0; i < 4; ++i) {
        const int idx4 = tid + i * NTHREADS;        // 0..1023 float4 chunks
        const int row  = idx4 >> 4;                 // 16 float4 per row
        const int col  = (idx4 & 15) * 4;
        const v4f v = *reinterpret_cast<const v4f*>(Ab + row * DIM + col);
        *reinterpret_cast<v4f*>(Xs + row * LSTR + col) = v;
    }
    if (tid < DIM) y0s[tid] = y0[(size_t)b * DIM + tid];
    __syncthreads();

    // ---- inf-norm of Omega = |h| * max_row sum|A_ij| ----
    if (tid < DIM) {
        float s = 0.0f;
        const float* r = Xs + tid * LSTR;
        #pragma unroll 8
        for (int c = 0; c < DIM; ++c) s += fabsf(r[c]);
        rowsum[tid] = s;
    }
    __syncthreads();

    float nrm = 0.0f;
    #pragma unroll 8
    for (int r = 0; r < DIM; ++r) nrm = fmaxf(nrm, rowsum[r]);
    nrm *= fabsf(hb);

    // ---- scaling exponent (uniform per block): ||X||/2^s <= theta ----
    const float theta = 0.25f;                      // deg-8 Taylor err ~ theta^9/9! ~ 1e-11
    int s = 0;
    if (nrm > theta) {
        s = (int)ceilf(log2f(nrm / theta));
        if (s < 0)  s = 0;
        if (s > 24) s = 24;
    }
    const float xscale = hb * exp2f((float)-s);     // fold h and 2^-s into one pass

    // ---- X *= h*2^-s (row-major, in place); C0 = I + X/8 col-major (Horner seed) ----
    #pragma unroll
    for (int i = 0; i < 4; ++i) {
        const int idx4 = tid + i * NTHREADS;
        const int row  = idx4 >> 4;
        const int col4 = (idx4 & 15) * 4;
        #pragma unroll
        for (int j = 0; j < 4; ++j) {
            const int col = col4 + j;
            const float x = Xs[row * LSTR + col] * xscale;
            Xs[row * LSTR + col] = x;
            C0[col * LSTR + row] = x * 0.125f + ((row == col) ? 1.0f : 0.0f);
        }
    }
    __syncthreads();

    // ---- Horner: T = I + (X @ T)/k, k = 7..1 ; last step also emits row-major T ----
    float* ccur = C0;
    float* cnxt = C1;
    #pragma unroll
    for (int k = 7; k >= 1; --k) {
        wmma_stage(Xs, ccur, cnxt, (k == 1) ? R0 : nullptr, 1.0f / (float)k, true);
        __syncthreads();
        float* t = ccur; ccur = cnxt; cnxt = t;
    }

    // ---- s squarings: T = T@T (uniform trip count); keep both reps live ----
    float* rcur = R0;
    float* rnxt = Xs;   // X no longer needed: becomes row-major pong
    for (int i = 0; i < s; ++i) {
        wmma_stage(rcur, ccur, cnxt, rnxt, 1.0f, false);
        __syncthreads();
        float* t = ccur; ccur = cnxt; cnxt = t;
        t = rcur; rcur = rnxt; rnxt = t;
    }

    // ---- y = T @ y0 : 4 partial dot products per row (row-major T), then reduce ----
    {
        const int row  = tid & 63;
        const int part = tid >> 6;                  // 0..3 -> 16 cols each
        const float* r = rcur + row * LSTR + part * 16;
        const float* v = y0s + part * 16;
        float acc = 0.0f;
        #pragma unroll
        for (int c = 0; c < 16; ++c) acc = fmaf(r[c], v[c], acc);
        psum[part * DIM + row] = acc;
    }
    __syncthreads();

    if (tid < DIM) {
        const float y = psum[tid] + psum[DIM + tid] + psum[2 * DIM + tid] + psum[3 * DIM + tid];
        yout[(size_t)b * DIM + tid] = y;
    }
}

extern "C" void kernel_launch(void* const* d_in, const int* in_sizes, int n_in,
                              void* d_out, int out_size, void* d_ws, size_t ws_size,
                              hipStream_t stream)
{
    const float* A  = (const float*)d_in[0];   // [8192, 64, 64] fp32
    const float* h  = (const float*)d_in[1];   // [8192] fp32
    const float* y0 = (const float*)d_in[2];   // [8192, 64] fp32
    float* yout     = (float*)d_out;           // [8192, 64] fp32

    const int B = in_sizes[1];                 // 8192 batches
    magnus2_expm_kernel<<<B, NTHREADS, 0, stream>>>(A, h, y0, yout);
}